// VectorQuantizer_23837068492940
// MI455X (gfx1250) — compile-verified
//
#include <hip/hip_runtime.h>

typedef float v2f __attribute__((ext_vector_type(2)));
typedef float v8f __attribute__((ext_vector_type(8)));

#define DIM 64
#define NCODES 512
#define NROWS (32 * 64 * 64)          // 131072 vectors
#define ROWS_PER_WAVE 16
#define NWAVES (NROWS / ROWS_PER_WAVE) // 8192
#define WAVES_PER_BLOCK 8
#define MAIN_BLOCKS (NWAVES / WAVES_PER_BLOCK) // 1024

// ws layout (float units):
//   [0, 512)          e_norm[k] = ||emb_k||^2
//   [512, 1024)       histogram (unsigned)
//   [1024, 1024+8192) per-wave SSE partials

// ---------------- prep: code norms + zero scratch ----------------
__global__ void vq_prep(const float* __restrict__ emb, float* __restrict__ ws) {
    int t = threadIdx.x; // 512 threads, one per code
    const float* e = emb + t * DIM;
    float s = 0.f;
#pragma unroll
    for (int i = 0; i < DIM; ++i) s += e[i] * e[i];
    ws[t] = s;
    ((unsigned*)ws)[512 + t] = 0u;
}

// ---------------- main: fused GEMM + argmin + outputs ----------------
__global__ __launch_bounds__(256) void vq_main(const float* __restrict__ x,
                                               const float* __restrict__ emb,
                                               float* __restrict__ ws,
                                               float* __restrict__ out) {
    const int lane   = threadIdx.x & 31;
    const int wave   = blockIdx.x * WAVES_PER_BLOCK + (threadIdx.x >> 5);
    const int row0   = wave * ROWS_PER_WAVE;
    const int lane16 = lane & 15;
    const int koff   = (lane >> 4) * 2; // WMMA f32 A/B layout: lanes 16-31 hold K+2,K+3

    // Preload A block (16 rows x 64 depth) in WMMA A layout: lane = row, 16 K-chunks.
    v2f a[16];
    const float* xr = x + (size_t)(row0 + lane16) * DIM;
#pragma unroll
    for (int c = 0; c < 16; ++c)
        a[c] = *(const v2f*)(xr + c * 4 + koff);

    float minval[8];
    int   minidx[8];
#pragma unroll
    for (int v = 0; v < 8; ++v) { minval[v] = 3.4e38f; minidx[v] = 0; }

    // Sweep 32 column tiles of 16 codes each; codebook (128 KB) stays cache-resident.
    for (int ct = 0; ct < NCODES / 16; ++ct) {
        const int   code = ct * 16 + lane16;
        const float en   = ws[code];                    // ||e||^2 for this lane's column
        const float* er  = emb + (size_t)code * DIM;
        v8f acc = {};
#pragma unroll
        for (int c = 0; c < 16; ++c) {
            v2f b = *(const v2f*)(er + c * 4 + koff);   // B tile chunk, same lane layout as A
            acc = __builtin_amdgcn_wmma_f32_16x16x4_f32(
                false, a[c], false, b, (short)0, acc, false, false);
        }
#pragma unroll
        for (int v = 0; v < 8; ++v) {
            float s = en - 2.0f * acc[v];               // ||x||^2 omitted: row-constant
            if (s < minval[v]) { minval[v] = s; minidx[v] = code; }
        }
    }

    // Cross-lane argmin within each 16-lane half (C layout: half = row group).
#pragma unroll
    for (int v = 0; v < 8; ++v) {
#pragma unroll
        for (int m = 1; m < 16; m <<= 1) {
            float ov = __shfl_xor(minval[v], m, 32);
            int   oi = __shfl_xor(minidx[v], m, 32);
            if (ov < minval[v] || (ov == minval[v] && oi < minidx[v])) {
                minval[v] = ov; minidx[v] = oi;
            }
        }
    }

    float*    out_q   = out + 1;
    float*    out_enc = out + 2 + (size_t)NROWS * DIM;
    unsigned* hist    = (unsigned*)ws + 512;

    float sse = 0.f;
#pragma unroll
    for (int m = 0; m < 16; ++m) {
        const int idx = __shfl(minidx[m & 7], (m < 8) ? 0 : 16, 32);
        const int row = row0 + m;

        // quantized = emb[idx]; accumulate SSE against x
        v2f q  = *(const v2f*)(emb + (size_t)idx * DIM + lane * 2);
        v2f xv = *(const v2f*)(x + (size_t)row * DIM + lane * 2);
        *(v2f*)(out_q + (size_t)row * DIM + lane * 2) = q;
        float d0 = q.x - xv.x, d1 = q.y - xv.y;
        sse += d0 * d0 + d1 * d1;

        if (lane == 0) atomicAdd(&hist[idx], 1u);

        // one-hot encodings row: 128 float4 stores spread over 32 lanes
        float* eb = out_enc + (size_t)row * NCODES;
#pragma unroll
        for (int t = 0; t < 4; ++t) {
            int c0 = (lane + 32 * t) * 4;
            float4 val;
            val.x = (c0 + 0 == idx) ? 1.f : 0.f;
            val.y = (c0 + 1 == idx) ? 1.f : 0.f;
            val.z = (c0 + 2 == idx) ? 1.f : 0.f;
            val.w = (c0 + 3 == idx) ? 1.f : 0.f;
            *(float4*)(eb + c0) = val;
        }
    }

    // Deterministic SSE: wave-reduce, one partial per wave (no float atomics).
#pragma unroll
    for (int m = 1; m < 32; m <<= 1) sse += __shfl_xor(sse, m, 32);
    if (lane == 0) ws[1024 + wave] = sse;
}

// ---------------- final: loss + perplexity ----------------
__global__ void vq_final(const float* __restrict__ ws, float* __restrict__ out) {
    __shared__ float red[512];
    const int t = threadIdx.x; // 512 threads

    const unsigned* hist = (const unsigned*)ws + 512;
    float p = (float)hist[t] / (float)NROWS;
    red[t] = p * logf(p + 1e-10f);
    __syncthreads();
    for (int s = 256; s > 0; s >>= 1) {
        if (t < s) red[t] += red[t + s];
        __syncthreads();
    }
    float H = red[0];
    __syncthreads();

    float ssum = 0.f;
    for (int i = 0; i < NWAVES / 512; ++i) ssum += ws[1024 + t + i * 512];
    red[t] = ssum;
    __syncthreads();
    for (int s = 256; s > 0; s >>= 1) {
        if (t < s) red[t] += red[t + s];
        __syncthreads();
    }
    if (t == 0) {
        float mse = red[0] / (float)((size_t)NROWS * DIM);
        out[0] = 1.25f * mse;                       // emb_loss + 0.25 * input_loss
        out[1 + (size_t)NROWS * DIM] = expf(-H);    // perplexity
    }
}

extern "C" void kernel_launch(void* const* d_in, const int* in_sizes, int n_in,
                              void* d_out, int out_size, void* d_ws, size_t ws_size,
                              hipStream_t stream) {
    const float* x   = (const float*)d_in[0];
    const float* emb = (const float*)d_in[1];
    float* out = (float*)d_out;
    float* ws  = (float*)d_ws;

    vq_prep<<<1, NCODES, 0, stream>>>(emb, ws);
    vq_main<<<MAIN_BLOCKS, 256, 0, stream>>>(x, emb, ws, out);
    vq_final<<<1, 512, 0, stream>>>(ws, out);
}